// PPIModel_67508295958926
// MI455X (gfx1250) — compile-verified
//
#include <hip/hip_runtime.h>
#include <math.h>

typedef __attribute__((ext_vector_type(2))) float v2f;
typedef __attribute__((ext_vector_type(8))) float v8f;

#define N_NODES  286
#define NPAD     288            // 18 tiles * 16 rows, bounds-check-free WMMA tiles
#define N_EDGES  16384
#define NTHREADS 512
#define NWAVES   (NTHREADS / 32)
#define NTILES   18

__global__ __launch_bounds__(NTHREADS)
void ppi_fused_kernel(const float* __restrict__ features,
                      const int*   __restrict__ src,
                      const int*   __restrict__ dst,
                      const float* __restrict__ W1, const float* __restrict__ b1,
                      const float* __restrict__ W2, const float* __restrict__ b2,
                      const float* __restrict__ fcW, const float* __restrict__ fcb,
                      float* __restrict__ out)
{
    __shared__ float xc[2 * NPAD];      // current (scaled) node features: GEMM input
    __shared__ float hh[2 * NPAD];      // h = xc @ W (per node)
    __shared__ float agg[2 * N_NODES];  // scatter accumulator
    __shared__ float rsdo[N_NODES];     // rsqrt(clip(out-degree))
    __shared__ float rsdi[N_NODES];     // rsqrt(clip(in-degree))
    __shared__ float wsm[12];           // [W1(4), b1(2), W2(4), b2(2)]
    __shared__ float red[NWAVES];       // per-wave partials for final dot

    const int tid  = threadIdx.x;
    const int lane = tid & 31;
    const int wave = tid >> 5;

    // ---- Stage weights/biases into LDS (one-time) ----
    if (tid < 4)                wsm[tid]     = W1[tid];
    else if (tid < 6)           wsm[tid]     = b1[tid - 4];
    else if (tid < 10)          wsm[tid]     = W2[tid - 6];
    else if (tid < 12)          wsm[tid]     = b2[tid - 10];

    // ---- Phase 0: degree counts (exact +1.0 adds -> deterministic) ----
    for (int i = tid; i < N_NODES; i += NTHREADS) { rsdo[i] = 0.f; rsdi[i] = 0.f; }
    __syncthreads();
    for (int e = tid; e < N_EDGES; e += NTHREADS) {
        atomicAdd(&rsdo[src[e]], 1.f);   // ds_add_f32
        atomicAdd(&rsdi[dst[e]], 1.f);
    }
    __syncthreads();
    for (int i = tid; i < N_NODES; i += NTHREADS) {
        rsdo[i] = rsqrtf(fmaxf(rsdo[i], 1.f));
        rsdi[i] = rsqrtf(fmaxf(rsdi[i], 1.f));
    }
    __syncthreads();

    // ---- Phase 1: layer-1 GEMM input: x * rsqrt(deg_out); zero the pad rows ----
    for (int i = tid; i < 2 * NPAD; i += NTHREADS)
        xc[i] = (i < 2 * N_NODES) ? features[i] * rsdo[i >> 1] : 0.f;
    __syncthreads();

    // ---- Two GraphConv layers ----
    for (int layer = 0; layer < 2; ++layer) {
        const int base = layer * 6;     // W at wsm[base..base+3], b at wsm[base+4..5]

        // zero the scatter accumulator (disjoint from WMMA arrays; same barrier covers both)
        for (int i = tid; i < 2 * N_NODES; i += NTHREADS) agg[i] = 0.f;

        // hh = xc @ W via V_WMMA_F32_16X16X4_F32, 16-row tiles, wave-uniform loop
        {
            const int  n  = lane & 15;
            const bool hi = (lane >= 16);   // lanes 16-31 carry K=2,3 -> zero padding
            const bool bv = (!hi && n < 2);
            v2f B;
            B.x = bv ? wsm[base + n]     : 0.f;   // K=0 row of W
            B.y = bv ? wsm[base + 2 + n] : 0.f;   // K=1 row of W
            for (int t = wave; t < NTILES; t += NWAVES) {
                const int m = t * 16 + n;          // < NPAD, always in-bounds
                const float2 xv = *(const float2*)(xc + 2 * m);   // ds_load_b64
                v2f A;
                A.x = hi ? 0.f : xv.x;             // K=0 (select, no exec dance)
                A.y = hi ? 0.f : xv.y;             // K=1
                v8f C = {0.f, 0.f, 0.f, 0.f, 0.f, 0.f, 0.f, 0.f};
                // (neg_a, A, neg_b, B, c_mod, C, reuse_a, reuse_b)
                C = __builtin_amdgcn_wmma_f32_16x16x4_f32(
                        false, A, false, B, (short)0, C, false, false);
                // D layout: VGPR r, lanes 0-15 -> (M=r, N=lane); lanes 16-31 -> (M=r+8, N=lane-16)
                if (n < 2) {
                    float* hp = hh + 2 * (t * 16 + (hi ? 8 : 0)) + n;
#pragma unroll
                    for (int r = 0; r < 8; ++r) hp[2 * r] = C[r];
                }
            }
        }
        __syncthreads();

        // scatter: agg[dst] += hh[src]
        for (int e = tid; e < N_EDGES; e += NTHREADS) {
            const int s = src[e], d = dst[e];
            const float2 h = *(const float2*)(hh + 2 * s);        // ds_load_b64
            atomicAdd(&agg[2 * d + 0], h.x);
            atomicAdd(&agg[2 * d + 1], h.y);
        }
        __syncthreads();

        // normalize + bias + ReLU; pre-scale by rsqrt(deg_out) for next layer's GEMM
        const float b0v = wsm[base + 4], b1v = wsm[base + 5];
        for (int i = tid; i < 2 * N_NODES; i += NTHREADS) {
            const int row = i >> 1;
            const int f   = i & 1;
            float v = fmaxf(agg[i] * rsdi[row] + (f ? b1v : b0v), 0.f);
            xc[i] = (layer == 0) ? v * rsdo[row] : v;   // pad rows untouched: stay zero
        }
        __syncthreads();
    }

    // ---- Final: sigmoid( flatten(x) . fc_W + fc_b ) ----
    float p = 0.f;
    for (int i = tid; i < 2 * N_NODES; i += NTHREADS) p += xc[i] * fcW[i];
#pragma unroll
    for (int off = 16; off > 0; off >>= 1) p += __shfl_down(p, off, 32);
    if (lane == 0) red[wave] = p;
    __syncthreads();
    if (tid == 0) {
        float s = fcb[0];
#pragma unroll
        for (int j = 0; j < NWAVES; ++j) s += red[j];
        out[0] = 1.f / (1.f + expf(-s));
    }
}

extern "C" void kernel_launch(void* const* d_in, const int* in_sizes, int n_in,
                              void* d_out, int out_size, void* d_ws, size_t ws_size,
                              hipStream_t stream) {
    (void)in_sizes; (void)n_in; (void)out_size; (void)d_ws; (void)ws_size;
    const float* features = (const float*)d_in[0];
    const int*   src      = (const int*)  d_in[1];
    const int*   dst      = (const int*)  d_in[2];
    const float* W1       = (const float*)d_in[3];
    const float* b1       = (const float*)d_in[4];
    const float* W2       = (const float*)d_in[5];
    const float* b2       = (const float*)d_in[6];
    const float* fcW      = (const float*)d_in[7];
    const float* fcb      = (const float*)d_in[8];
    ppi_fused_kernel<<<1, NTHREADS, 0, stream>>>(features, src, dst,
                                                 W1, b1, W2, b2, fcW, fcb,
                                                 (float*)d_out);
}